// TalkingHeadAttention_57535381897300
// MI455X (gfx1250) — compile-verified
//
#include <hip/hip_runtime.h>

// ---------------- problem constants ----------------
#define BATCH   4
#define SEQ     1024
#define CDIM    768
#define NHEADS  12
#define DHEAD   64
#define C3      (3 * CDIM)          // 2304
#define QSCALE  0.125f              // 64^-0.5

typedef __attribute__((ext_vector_type(2))) float v2f;
typedef __attribute__((ext_vector_type(8))) float v8f;

#define WMMA_F32(a, b, c) \
  __builtin_amdgcn_wmma_f32_16x16x4_f32(false, (a), false, (b), (short)0, (c), false, false)

// ---------------------------------------------------------------------------
// Per-wave 16x16 fp32 tile GEMM using V_WMMA_F32_16X16X4_F32.
// ISA layouts (CDNA5 ISA §7.12.2):
//   A 16x4 : lane<16 -> row=lane,   v0=K0, v1=K1 ; lane>=16 -> row=lane-16, v0=K2, v1=K3
//   B 4x16 : lane<16 -> col=lane,   v0=K0, v1=K1 ; lane>=16 -> col=lane-16, v0=K2, v1=K3
//   C/D    : vgpr r, lane L -> row = r + (L>=16 ? 8:0), col = L%16
// Dual accumulators break the serial D->C dependency chain in the XDL pipe.
// ---------------------------------------------------------------------------

// C = A[row0.., :K] * B[:K, col0..]          (A MxK row-major, B KxN row-major)
__device__ __forceinline__ v8f wmma_tile_nn(const float* __restrict__ A, int lda,
                                            const float* __restrict__ Bm, int ldb,
                                            int row0, int col0, int K, int lane) {
  const int half = lane >> 4, l16 = lane & 15;
  const float* arow = A + (size_t)(row0 + l16) * lda;
  const float* bcol = Bm + col0 + l16;
  v8f c0 = {}, c1 = {};
  for (int kk = 0; kk < K; kk += 8) {
    {
      const int ka = kk + half * 2;
      v2f a = *(const v2f*)(arow + ka);        // 8B aligned: lda, ka even
      v2f b;
      b[0] = bcol[(size_t)ka * ldb];
      b[1] = bcol[(size_t)(ka + 1) * ldb];
      c0 = WMMA_F32(a, b, c0);
    }
    {
      const int ka = kk + 4 + half * 2;
      v2f a = *(const v2f*)(arow + ka);
      v2f b;
      b[0] = bcol[(size_t)ka * ldb];
      b[1] = bcol[(size_t)(ka + 1) * ldb];
      c1 = WMMA_F32(a, b, c1);
    }
  }
  return c0 + c1;
}

// C = A[row0.., :K] * Bt[col0.., :K]^T       (Bt NxK row-major -> B = Bt^T)
__device__ __forceinline__ v8f wmma_tile_nt(const float* __restrict__ A, int lda,
                                            const float* __restrict__ Bt, int ldbt,
                                            int row0, int col0, int K, int lane) {
  const int half = lane >> 4, l16 = lane & 15;
  const float* arow = A + (size_t)(row0 + l16) * lda;
  const float* brow = Bt + (size_t)(col0 + l16) * ldbt;
  v8f c0 = {}, c1 = {};
  for (int kk = 0; kk < K; kk += 8) {
    {
      const int ka = kk + half * 2;
      v2f a = *(const v2f*)(arow + ka);
      v2f b = *(const v2f*)(brow + ka);
      c0 = WMMA_F32(a, b, c0);
    }
    {
      const int ka = kk + 4 + half * 2;
      v2f a = *(const v2f*)(arow + ka);
      v2f b = *(const v2f*)(brow + ka);
      c1 = WMMA_F32(a, b, c1);
    }
  }
  return c0 + c1;
}

// ---------------------------------------------------------------------------
// K1: qkv = x @ W_qkv + b ; scatter into q(scaled)/k/v as [B,H,N,d]
// tiles = (4096/16)*(2304/16) = 256*144 ; 8 waves/block
// ---------------------------------------------------------------------------
__global__ __launch_bounds__(256) void k1_qkv(const float* __restrict__ x,
                                              const float* __restrict__ Wqkv,
                                              const float* __restrict__ bqkv,
                                              float* __restrict__ qb,
                                              float* __restrict__ kb,
                                              float* __restrict__ vb) {
  const int lane = threadIdx.x & 31;
  const int tile = blockIdx.x * 8 + (threadIdx.x >> 5);
  const int ct = tile % (C3 / 16);
  const int rt = tile / (C3 / 16);
  v8f c = wmma_tile_nn(x, CDIM, Wqkv, C3, rt * 16, ct * 16, CDIM, lane);
  const int half = lane >> 4, l16 = lane & 15;
  const int col = ct * 16 + l16;
  const int t = col / CDIM;             // 0=q 1=k 2=v
  const int h = (col % CDIM) / DHEAD;
  const int dd = col % DHEAD;
  const float bias = bqkv[col];
  const float scl = (t == 0) ? QSCALE : 1.0f;
  float* dst = (t == 0) ? qb : (t == 1) ? kb : vb;
  for (int r = 0; r < 8; ++r) {
    const int g = rt * 16 + r + half * 8;          // b*SEQ + n
    const int b = g >> 10, n = g & (SEQ - 1);
    dst[((size_t)(b * NHEADS + h) * SEQ + n) * DHEAD + dd] = (c[r] + bias) * scl;
  }
}

// ---------------------------------------------------------------------------
// K2: logits = q @ k^T fused with first talking-heads mix -> d_out.attn
// 12 waves (one per head) compute 16x16 tiles, LDS exchange, wave k mixes.
// grid = B * 64 * 64
// ---------------------------------------------------------------------------
__global__ __launch_bounds__(384) void k2_logits_mix(const float* __restrict__ qb,
                                                     const float* __restrict__ kb,
                                                     const float* __restrict__ Wl,
                                                     const float* __restrict__ bl,
                                                     float* __restrict__ attn) {
  __shared__ float sW[NHEADS * NHEADS];
  __shared__ float sb[NHEADS];
  __shared__ float tiles[NHEADS * 256];
  if (threadIdx.x < NHEADS * NHEADS) sW[threadIdx.x] = Wl[threadIdx.x];
  if (threadIdx.x < NHEADS)          sb[threadIdx.x] = bl[threadIdx.x];

  const int lane = threadIdx.x & 31;
  const int h = threadIdx.x >> 5;                 // wave id == head id
  int bx = blockIdx.x;
  const int mt = bx & 63;  bx >>= 6;
  const int nt = bx & 63;  bx >>= 6;
  const int b  = bx;

  const float* q = qb + (size_t)(b * NHEADS + h) * SEQ * DHEAD;
  const float* k = kb + (size_t)(b * NHEADS + h) * SEQ * DHEAD;
  v8f c = wmma_tile_nt(q, DHEAD, k, DHEAD, nt * 16, mt * 16, DHEAD, lane);
  for (int r = 0; r < 8; ++r) tiles[h * 256 + r * 32 + lane] = c[r];
  __syncthreads();

  const int k2 = h;                               // wave id == output head id
  const int half = lane >> 4, l16 = lane & 15;
  float outr[8];
  for (int r = 0; r < 8; ++r) outr[r] = sb[k2];
  for (int hh = 0; hh < NHEADS; ++hh) {
    const float w = sW[hh * NHEADS + k2];
    for (int r = 0; r < 8; ++r) outr[r] += tiles[hh * 256 + r * 32 + lane] * w;
  }
  float* dst = attn + ((size_t)(b * NHEADS + k2) * SEQ + nt * 16) * SEQ + mt * 16;
  for (int r = 0; r < 8; ++r)
    dst[(size_t)(r + half * 8) * SEQ + l16] = outr[r];
}

// ---------------------------------------------------------------------------
// K3: in-place row softmax over last axis. grid = B*H*N rows of 1024.
// ---------------------------------------------------------------------------
__global__ __launch_bounds__(256) void k3_softmax(float* __restrict__ attn) {
  __shared__ float red[256];
  float* p = attn + (size_t)blockIdx.x * SEQ;
  const int t = threadIdx.x;
  float v[4], m = -3.402823466e38f;
  for (int i = 0; i < 4; ++i) { v[i] = p[t + i * 256]; m = fmaxf(m, v[i]); }
  red[t] = m; __syncthreads();
  for (int s = 128; s > 0; s >>= 1) {
    if (t < s) red[t] = fmaxf(red[t], red[t + s]);
    __syncthreads();
  }
  m = red[0]; __syncthreads();
  float sum = 0.f;
  for (int i = 0; i < 4; ++i) { v[i] = __expf(v[i] - m); sum += v[i]; }
  red[t] = sum; __syncthreads();
  for (int s = 128; s > 0; s >>= 1) {
    if (t < s) red[t] += red[t + s];
    __syncthreads();
  }
  const float inv = 1.0f / red[0];
  for (int i = 0; i < 4; ++i) p[t + i * 256] = v[i] * inv;
}

// ---------------------------------------------------------------------------
// K4: second talking-heads mix, in place. One thread owns one (b,n,m) slot
// across all 12 heads -> race-free in-place update. grid = B*N*N/256.
// ---------------------------------------------------------------------------
__global__ __launch_bounds__(256) void k4_mix2(const float* __restrict__ Ww,
                                               const float* __restrict__ bw,
                                               float* __restrict__ attn) {
  __shared__ float sW[NHEADS * NHEADS];
  __shared__ float sb[NHEADS];
  if (threadIdx.x < NHEADS * NHEADS) sW[threadIdx.x] = Ww[threadIdx.x];
  if (threadIdx.x < NHEADS)          sb[threadIdx.x] = bw[threadIdx.x];
  __syncthreads();
  const size_t NN = (size_t)SEQ * SEQ;
  const size_t t  = (size_t)blockIdx.x * 256 + threadIdx.x;
  const size_t b  = t / NN, r = t % NN;
  float* base = attn + b * NHEADS * NN + r;
  float s[NHEADS];
  for (int k = 0; k < NHEADS; ++k) s[k] = base[k * NN];
  for (int k2 = 0; k2 < NHEADS; ++k2) {
    float o = sb[k2];
    for (int k = 0; k < NHEADS; ++k) o += s[k] * sW[k * NHEADS + k2];
    base[k2 * NN] = o;
  }
}

// ---------------------------------------------------------------------------
// K5: ctx[b,n,h*64+d] = attn2[b,h] @ v[b,h]   (1024x1024 @ 1024x64 per head)
// One wave computes a full 16(n) x 64(d) strip: the A-fragment (attn, the
// 201MB tensor) is loaded ONCE per K-step and feeds 4 independent WMMA
// chains; the 4 B-fragments consume each K-row of v (256B) fully coalesced.
// wave-strips = B*H*64 = 3072 ; 8 waves/block -> 384 blocks.
// ---------------------------------------------------------------------------
__global__ __launch_bounds__(256) void k5_av(const float* __restrict__ attn,
                                             const float* __restrict__ vb,
                                             float* __restrict__ ctx) {
  const int lane = threadIdx.x & 31;
  int strip = blockIdx.x * 8 + (threadIdx.x >> 5);
  const int nt = strip & 63;  strip >>= 6;
  const int h  = strip % NHEADS;
  const int b  = strip / NHEADS;
  const float* A = attn + (size_t)(b * NHEADS + h) * SEQ * SEQ;
  const float* V = vb   + (size_t)(b * NHEADS + h) * SEQ * DHEAD;

  const int half = lane >> 4, l16 = lane & 15;
  const float* arow = A + (size_t)(nt * 16 + l16) * SEQ;
  const float* vcol = V + l16;
  v8f c0 = {}, c1 = {}, c2 = {}, c3 = {};
  for (int kk = 0; kk < SEQ; kk += 4) {
    const int ka = kk + half * 2;
    v2f a = *(const v2f*)(arow + ka);
    const float* vr0 = vcol + (size_t)ka * DHEAD;
    const float* vr1 = vr0 + DHEAD;
    v2f b0, b1, b2, b3;
    b0[0] = vr0[0];  b0[1] = vr1[0];
    b1[0] = vr0[16]; b1[1] = vr1[16];
    b2[0] = vr0[32]; b2[1] = vr1[32];
    b3[0] = vr0[48]; b3[1] = vr1[48];
    c0 = WMMA_F32(a, b0, c0);
    c1 = WMMA_F32(a, b1, c1);
    c2 = WMMA_F32(a, b2, c2);
    c3 = WMMA_F32(a, b3, c3);
  }
  for (int r = 0; r < 8; ++r) {
    const int n = nt * 16 + r + half * 8;
    float* dst = ctx + (size_t)(b * SEQ + n) * CDIM + h * DHEAD + l16;
    dst[0]  = c0[r];
    dst[16] = c1[r];
    dst[32] = c2[r];
    dst[48] = c3[r];
  }
}

// ---------------------------------------------------------------------------
// K6: out = ctx @ W_proj + b_proj. tiles = 256*48 = 12288 ; 8 waves/block
// ---------------------------------------------------------------------------
__global__ __launch_bounds__(256) void k6_proj(const float* __restrict__ ctx,
                                               const float* __restrict__ Wp,
                                               const float* __restrict__ bp,
                                               float* __restrict__ out) {
  const int lane = threadIdx.x & 31;
  const int tile = blockIdx.x * 8 + (threadIdx.x >> 5);
  const int ct = tile % (CDIM / 16);
  const int rt = tile / (CDIM / 16);
  v8f c = wmma_tile_nn(ctx, CDIM, Wp, CDIM, rt * 16, ct * 16, CDIM, lane);
  const int half = lane >> 4, l16 = lane & 15;
  const int col = ct * 16 + l16;
  const float bias = bp[col];
  for (int r = 0; r < 8; ++r)
    out[(size_t)(rt * 16 + r + half * 8) * CDIM + col] = c[r] + bias;
}

// ---------------------------------------------------------------------------
extern "C" void kernel_launch(void* const* d_in, const int* in_sizes, int n_in,
                              void* d_out, int out_size, void* d_ws, size_t ws_size,
                              hipStream_t stream) {
  const float* x     = (const float*)d_in[0];
  const float* Wqkv  = (const float*)d_in[1];
  const float* bqkv  = (const float*)d_in[2];
  const float* Wl    = (const float*)d_in[3];
  const float* bl    = (const float*)d_in[4];
  const float* Ww    = (const float*)d_in[5];
  const float* bw    = (const float*)d_in[6];
  const float* Wproj = (const float*)d_in[7];
  const float* bproj = (const float*)d_in[8];

  // d_out: [out (B*N*C)] ++ [attn (B*H*N*N)]
  float* out  = (float*)d_out;
  float* attn = (float*)d_out + (size_t)BATCH * SEQ * CDIM;

  // workspace: q | k | v | ctx  (each B*N*C floats = 12 MB, total 48 MB)
  const size_t SLAB = (size_t)BATCH * NHEADS * SEQ * DHEAD;   // == B*N*C
  float* qb  = (float*)d_ws;
  float* kb  = qb + SLAB;
  float* vb  = kb + SLAB;
  float* ctx = vb + SLAB;

  // K1: 256*144 tiles / 8 waves
  k1_qkv<<<(256 * 144) / 8, 256, 0, stream>>>(x, Wqkv, bqkv, qb, kb, vb);
  // K2: B*64*64 blocks, 12 waves each
  k2_logits_mix<<<BATCH * 64 * 64, 384, 0, stream>>>(qb, kb, Wl, bl, attn);
  // K3: one block per row
  k3_softmax<<<BATCH * NHEADS * SEQ, 256, 0, stream>>>(attn);
  // K4: B*N*N threads
  k4_mix2<<<(BATCH * SEQ * SEQ) / 256, 256, 0, stream>>>(Ww, bw, attn);
  // K5: 3072 strips / 8 waves
  k5_av<<<3072 / 8, 256, 0, stream>>>(attn, vb, ctx);
  // K6: 12288 tiles / 8 waves
  k6_proj<<<12288 / 8, 256, 0, stream>>>(ctx, Wproj, bproj, out);
}